// LocalLocalContrastiveLoss_38439957299269
// MI455X (gfx1250) — compile-verified
//
#include <hip/hip_runtime.h>
#include <math.h>

// ---------------------------------------------------------------------------
// LocalLocalContrastiveLoss on MI455X (gfx1250)
//   B=16, L=256, D=256, K=16384  ->  N = 4096 anchors, 20480 logit columns
//   Fused matmul + online logsumexp using V_WMMA_F32_16X16X4_F32 (full fp32).
//   sched_group_barrier forces a LOADx24 / (WMMAx8, LOADx8)* pipeline so the
//   B-fragment L2 latency overlaps the XDL pipe instead of serializing.
// ---------------------------------------------------------------------------

typedef __attribute__((ext_vector_type(2))) float v2f;
typedef __attribute__((ext_vector_type(8))) float v8f;

namespace {
constexpr int   Dq     = 256;           // feature dim
constexpr int   Nq     = 4096;          // B*L anchors
constexpr int   Kq     = 16384;         // memory queue rows
constexpr int   Lq     = 256;           // sequence length (fixed by reference)
constexpr int   NCOL   = Nq + Kq;       // 20480 logit columns
constexpr int   NTILE  = NCOL / 16;     // 1280 column tiles
constexpr float INV_T  = 1.0f / 0.07f;  // 14.2857...
}

#define WMMA_F32(acc, aa, bb)                                                  \
  acc = __builtin_amdgcn_wmma_f32_16x16x4_f32(false, (aa), false, (bb),        \
                                              (short)0, (acc), false, false)

// sched_group_barrier masks
#define SGB      __builtin_amdgcn_sched_group_barrier
#define SGB_WMMA 0x008  // MFMA / WMMA
#define SGB_VMRD 0x020  // VMEM read

// ----------------------------- Kernel 1 ------------------------------------
// One workgroup (8 waves) per 16-anchor-row tile. Waves stride over the 1280
// 16-column tiles. A fragments (16 rows x 256 K) live in 128 VGPRs per lane.
__global__ __launch_bounds__(256) void lse_wmma_kernel(
    const float* __restrict__ z,      // [4096, 256]
    const float* __restrict__ memq,   // [16384, 256]
    float* __restrict__ lse_out)      // [4096]
{
  const int r0   = blockIdx.x * 16;     // anchor row base
  const int tid  = threadIdx.x;
  const int lane = tid & 31;
  const int wave = tid >> 5;            // 0..7
  const int lrow = lane & 15;           // row (A) / col (B) within tile
  const int khi  = (lane >> 4) * 2;     // K sub-offset: 0 (lanes 0-15) / 2 (16-31)

  // Resident A fragments: 64 chunks of K=4, each a float2 per lane.
  v2f a[64];
  {
    const float* arow = z + (size_t)(r0 + lrow) * Dq;
#pragma unroll
    for (int kc = 0; kc < 64; ++kc)
      a[kc] = *(const v2f*)(arow + kc * 4 + khi);
  }

  // Online softmax state: 8 row-slots per lane (row = v + 8*(lane>=16)),
  // each slot streams one column (lane&15) per visited tile.
  float m[8], s[8];
#pragma unroll
  for (int v = 0; v < 8; ++v) { m[v] = -1e30f; s[v] = 0.0f; }

  for (int t = wave; t < NTILE; t += 8) {
    const int c0 = t * 16;
    const float* brow = (c0 < Nq)
        ? (z    + (size_t)(c0      + lrow) * Dq)
        : (memq + (size_t)(c0 - Nq + lrow) * Dq);

    v8f acc = {};        // zero C accumulator
    v2f b[3][8];         // 3-deep rotating B-fragment buffers (8 K-chunks each)

    // Prologue: prefetch groups 0..2 (24 fragments, 48 VGPRs in flight).
#pragma unroll
    for (int g = 0; g < 3; ++g)
#pragma unroll
      for (int j = 0; j < 8; ++j)
        b[g][j] = *(const v2f*)(brow + g * 32 + j * 4 + khi);

#pragma unroll
    for (int g = 0; g < 8; ++g) {
#pragma unroll
      for (int j = 0; j < 8; ++j) WMMA_F32(acc, a[g * 8 + j], b[g % 3][j]);
      if (g + 3 < 8) {                       // refill the buffer just drained
#pragma unroll
        for (int j = 0; j < 8; ++j)
          b[g % 3][j] = *(const v2f*)(brow + (g + 3) * 32 + j * 4 + khi);
      }
    }

    // Force the pipelined schedule: 24 loads up front, then alternate
    // 8 WMMAs / 8 loads, tail of 24 WMMAs. Loads stay >=2 groups ahead.
    SGB(SGB_VMRD, 24, 0);
#pragma unroll
    for (int g = 0; g < 5; ++g) {
      SGB(SGB_WMMA, 8, 0);
      SGB(SGB_VMRD, 8, 0);
    }
    SGB(SGB_WMMA, 24, 0);

    const bool diag = (c0 == r0);  // only tile that can contain self-sims
#pragma unroll
    for (int v = 0; v < 8; ++v) {
      float x = acc[v] * INV_T;
      const int rowm = v + (khi << 2);           // v + 8*(lane>=16)
      if (diag && rowm == lrow) x = -INFINITY;   // mask self-similarity
      const float mo = m[v];
      const float mn = fmaxf(mo, x);
      s[v] = s[v] * __expf(mo - mn) + __expf(x - mn);
      m[v] = mn;
    }
  }

  // Merge the 16 column-lanes inside each half (xor 1,2,4,8 never crosses bit4).
#pragma unroll
  for (int off = 1; off < 16; off <<= 1) {
#pragma unroll
    for (int v = 0; v < 8; ++v) {
      const float mo2 = __shfl_xor(m[v], off, 32);
      const float so2 = __shfl_xor(s[v], off, 32);
      const float mn  = fmaxf(m[v], mo2);
      s[v] = s[v] * __expf(m[v] - mn) + so2 * __expf(mo2 - mn);
      m[v] = mn;
    }
  }

  // Merge the 8 waves through LDS, then emit lse per row.
  __shared__ float red_m[16][8];
  __shared__ float red_s[16][8];
  if (lrow == 0) {
    const int rbase = (lane >> 4) * 8;
#pragma unroll
    for (int v = 0; v < 8; ++v) {
      red_m[rbase + v][wave] = m[v];
      red_s[rbase + v][wave] = s[v];
    }
  }
  __syncthreads();
  if (tid < 16) {
    float M = red_m[tid][0], S = red_s[tid][0];
#pragma unroll
    for (int w = 1; w < 8; ++w) {
      const float mo2 = red_m[tid][w];
      const float so2 = red_s[tid][w];
      const float mn  = fmaxf(M, mo2);
      S = S * __expf(M - mn) + so2 * __expf(mo2 - mn);
      M = mn;
    }
    lse_out[r0 + tid] = M + __logf(S);
  }
}

// ----------------------------- Kernel 2 ------------------------------------
// One block; deterministic tree reduction (no float atomics -> bitwise
// reproducible across graph replays). pos_sim recomputed directly (cheap).
__global__ __launch_bounds__(256) void loss_reduce_kernel(
    const float* __restrict__ z,
    const float* __restrict__ lse,
    float* __restrict__ out)
{
  const int tid = threadIdx.x;
  float local = 0.0f;
  for (int i = tid; i < Nq - 1; i += 256) {
    if ((i % Lq) == (Lq - 1)) continue;  // last-in-sequence anchors invalid
    const float* pa = z + (size_t)i * Dq;
    const float* pb = pa + Dq;
    float dot = 0.0f;
    for (int k = 0; k < Dq; k += 4) {
      const float4 x = *(const float4*)(pa + k);
      const float4 y = *(const float4*)(pb + k);
      dot += x.x * y.x + x.y * y.y + x.z * y.z + x.w * y.w;
    }
    local += lse[i] - dot * INV_T;
  }
  __shared__ float buf[256];
  buf[tid] = local;
  __syncthreads();
  for (int off = 128; off > 0; off >>= 1) {
    if (tid < off) buf[tid] += buf[tid + off];
    __syncthreads();
  }
  if (tid == 0) out[0] = buf[0] / (float)(16 * (Lq - 1));  // B*(L-1) = 4080
}

// ----------------------------- Launcher ------------------------------------
extern "C" void kernel_launch(void* const* d_in, const int* in_sizes, int n_in,
                              void* d_out, int out_size, void* d_ws, size_t ws_size,
                              hipStream_t stream) {
  const float* z    = (const float*)d_in[0];  // z_t          [16,256,256]
  // d_in[1] = va_values -- unused (faithful to reference: it never affects loss)
  const float* memq = (const float*)d_in[2];  // memory_queue [16384,256]
  float* lse = (float*)d_ws;                  // 4096 floats of scratch
  float* out = (float*)d_out;                 // scalar loss

  lse_wmma_kernel<<<Nq / 16, 256, 0, stream>>>(z, memq, lse);
  loss_reduce_kernel<<<1, 256, 0, stream>>>(z, lse, out);
}